// SpacecraftGNN_6098853560889
// MI455X (gfx1250) — compile-verified
//
#include <hip/hip_runtime.h>
#include <math.h>

#define HD 16

typedef float v2f __attribute__((ext_vector_type(2)));
typedef float v8f __attribute__((ext_vector_type(8)));

// ---------------- degree / normalization ----------------

__global__ void k_init_deg(float* __restrict__ deg, int N) {
  int n = blockIdx.x * blockDim.x + threadIdx.x;
  if (n < N) deg[n] = 1.0f;  // self-loop contributes 1 to every node's degree
}

__global__ void k_deg_accum(const int* __restrict__ dst, float* __restrict__ deg, int E) {
  int e = blockIdx.x * blockDim.x + threadIdx.x;
  if (e < E) atomicAdd(&deg[dst[e]], 1.0f);
}

__global__ void k_dinv(float* __restrict__ deg, int N) {
  int n = blockIdx.x * blockDim.x + threadIdx.x;
  if (n < N) {
    float d = deg[n];
    deg[n] = (d > 0.0f) ? rsqrtf(d) : 0.0f;  // deg >= 1 always, but match reference
  }
}

// ---------------- conv1 pre: g = (x @ W1) * dinv ; acc = g (self-loop) -------

__global__ void k_conv1_pre(const float* __restrict__ x, const float* __restrict__ W1,
                            const float* __restrict__ dinv, float* __restrict__ g,
                            float* __restrict__ acc, int N) {
  int n = blockIdx.x * blockDim.x + threadIdx.x;
  if (n >= N) return;
  float s = x[n] * dinv[n];
  float4* g4 = (float4*)(g + (size_t)n * HD);
  float4* a4 = (float4*)(acc + (size_t)n * HD);
  const float4* w4 = (const float4*)W1;
#pragma unroll
  for (int q = 0; q < 4; ++q) {
    float4 w = w4[q];
    float4 v = make_float4(s * w.x, s * w.y, s * w.z, s * w.w);
    g4[q] = v;
    a4[q] = v;
  }
}

// ---------------- edge scatter: acc[dst] += g[src] (4 lanes per edge) -------

__global__ void k_scatter(const int* __restrict__ src, const int* __restrict__ dst,
                          const float* __restrict__ g, float* __restrict__ acc, int E) {
  long t = (long)blockIdx.x * blockDim.x + threadIdx.x;
  int e = (int)(t >> 2);
  if (e >= E) return;
  int q = (int)(t & 3);
  int s = src[e];
  int d = dst[e];
  float4 v = *(const float4*)(g + (size_t)s * HD + q * 4);
  float* a = acc + (size_t)d * HD + q * 4;
  atomicAdd(a + 0, v.x);
  atomicAdd(a + 1, v.y);
  atomicAdd(a + 2, v.z);
  atomicAdd(a + 3, v.w);
}

// ---------------- conv1 post: h1 = relu(dinv * acc + b1) --------------------

__global__ void k_post_relu(const float* __restrict__ acc, const float* __restrict__ dinv,
                            const float* __restrict__ b, float* __restrict__ hout, int N) {
  long t = (long)blockIdx.x * blockDim.x + threadIdx.x;
  int n = (int)(t >> 2);
  if (n >= N) return;
  int q = (int)(t & 3);
  float di = dinv[n];
  float4 a = *(const float4*)(acc + (size_t)n * HD + q * 4);
  float4 bb = ((const float4*)b)[q];
  float4 r;
  r.x = fmaxf(fmaf(di, a.x, bb.x), 0.0f);
  r.y = fmaxf(fmaf(di, a.y, bb.y), 0.0f);
  r.z = fmaxf(fmaf(di, a.z, bb.z), 0.0f);
  r.w = fmaxf(fmaf(di, a.w, bb.w), 0.0f);
  *(float4*)(hout + (size_t)n * HD + q * 4) = r;
}

// ---------------- conv2 pre via WMMA f32 16x16x4 ----------------------------
// One wave per 16-node tile: D(16x16) = A(16x16 node tile) x B(=W2), then
// g = D * dinv[row], acc = g (self-loop init).
// VGPR layouts per CDNA5 ISA 7.12.2:
//   A (16x4/chunk): lane L<16 holds row L, K = 4c+{0,1}; lanes 16-31: K = 4c+{2,3}
//   B (4x16/chunk): lane L<16 holds col L, K = 4c+{0,1}; lanes 16-31: K = 4c+{2,3}
//   C/D: lane L holds col (L&15); VGPR v = row v + 8*(L>>4)

__global__ void __launch_bounds__(256)
k_conv2_pre_wmma(const float* __restrict__ h1, const float* __restrict__ W2,
                 const float* __restrict__ dinv, float* __restrict__ g,
                 float* __restrict__ acc, int n_tiles) {
  int wave = (int)(((long)blockIdx.x * blockDim.x + threadIdx.x) >> 5);
  if (wave >= n_tiles) return;  // wave-uniform: EXEC stays all-1s for WMMA
  int lane = threadIdx.x & 31;
  int col = lane & 15;
  int hi = lane >> 4;
  int base = wave * 16;

  const float* arow = h1 + (size_t)(base + col) * HD;  // A row m = col
  v8f d = {0.f, 0.f, 0.f, 0.f, 0.f, 0.f, 0.f, 0.f};
#pragma unroll
  for (int c = 0; c < 4; ++c) {
    v2f a, b;
    a[0] = arow[4 * c + 2 * hi + 0];
    a[1] = arow[4 * c + 2 * hi + 1];
    b[0] = W2[(4 * c + 2 * hi + 0) * HD + col];
    b[1] = W2[(4 * c + 2 * hi + 1) * HD + col];
    d = __builtin_amdgcn_wmma_f32_16x16x4_f32(false, a, false, b, (short)0, d,
                                              false, false);
  }
#pragma unroll
  for (int v = 0; v < 8; ++v) {
    int row = base + v + 8 * hi;
    float val = d[v] * dinv[row];
    g[(size_t)row * HD + col] = val;
    acc[(size_t)row * HD + col] = val;
  }
}

// Scalar tail for N % 16 != 0 (not launched for N = 100000)
__global__ void k_conv2_pre_tail(const float* __restrict__ h1, const float* __restrict__ W2,
                                 const float* __restrict__ dinv, float* __restrict__ g,
                                 float* __restrict__ acc, int N, int start) {
  int n = start + blockIdx.x * blockDim.x + threadIdx.x;
  if (n >= N) return;
  float di = dinv[n];
  for (int j = 0; j < HD; ++j) {
    float s = 0.0f;
    for (int k = 0; k < HD; ++k) s += h1[(size_t)n * HD + k] * W2[k * HD + j];
    s *= di;
    g[(size_t)n * HD + j] = s;
    acc[(size_t)n * HD + j] = s;
  }
}

// ---------------- conv2 post fused with final head --------------------------
// h2 = relu(dinv*acc + b2); out = tanh(h2 @ Wf + bf)

__global__ void k_final(const float* __restrict__ acc, const float* __restrict__ dinv,
                        const float* __restrict__ b2, const float* __restrict__ Wf,
                        const float* __restrict__ bf, float* __restrict__ out, int N) {
  int n = blockIdx.x * blockDim.x + threadIdx.x;
  if (n >= N) return;
  float di = dinv[n];
  float sum = bf[0];
#pragma unroll
  for (int q = 0; q < 4; ++q) {
    float4 a = *(const float4*)(acc + (size_t)n * HD + q * 4);
    float4 b = ((const float4*)b2)[q];
    float4 w = ((const float4*)Wf)[q];
    sum += fmaxf(fmaf(di, a.x, b.x), 0.0f) * w.x;
    sum += fmaxf(fmaf(di, a.y, b.y), 0.0f) * w.y;
    sum += fmaxf(fmaf(di, a.z, b.z), 0.0f) * w.z;
    sum += fmaxf(fmaf(di, a.w, b.w), 0.0f) * w.w;
  }
  out[n] = tanhf(sum);
}

// ---------------- host launcher ---------------------------------------------

extern "C" void kernel_launch(void* const* d_in, const int* in_sizes, int n_in,
                              void* d_out, int out_size, void* d_ws, size_t ws_size,
                              hipStream_t stream) {
  const float* x  = (const float*)d_in[0];
  const int*   ei = (const int*)d_in[1];
  const float* W1 = (const float*)d_in[2];
  const float* b1 = (const float*)d_in[3];
  const float* W2 = (const float*)d_in[4];
  const float* b2 = (const float*)d_in[5];
  const float* Wf = (const float*)d_in[6];
  const float* bf = (const float*)d_in[7];
  float* out = (float*)d_out;

  int N = in_sizes[0];       // x is [N,1]
  int E = in_sizes[1] / 2;   // edge_index is [2,E]
  const int* src = ei;
  const int* dst = ei + E;

  float* ws   = (float*)d_ws;
  float* dinv = ws;                        // N floats (deg, then dinv in place)
  float* g    = dinv + N;                  // N*16
  float* acc  = g + (size_t)N * HD;        // N*16
  float* h1   = acc + (size_t)N * HD;      // N*16

  dim3 blk(256);
  int gn = (N + 255) / 256;
  int ge = (E + 255) / 256;
  long sc = (long)E * 4;
  int gs = (int)((sc + 255) / 256);
  long p4 = (long)N * 4;
  int gp = (int)((p4 + 255) / 256);

  // normalization
  k_init_deg<<<gn, blk, 0, stream>>>(dinv, N);
  k_deg_accum<<<ge, blk, 0, stream>>>(dst, dinv, E);
  k_dinv<<<gn, blk, 0, stream>>>(dinv, N);

  // conv1
  k_conv1_pre<<<gn, blk, 0, stream>>>(x, W1, dinv, g, acc, N);
  k_scatter<<<gs, blk, 0, stream>>>(src, dst, g, acc, E);
  k_post_relu<<<gp, blk, 0, stream>>>(acc, dinv, b1, h1, N);

  // conv2 dense transform via WMMA
  int n_tiles = N / 16;
  if (n_tiles > 0) {
    int gw = (n_tiles * 32 + 255) / 256;
    k_conv2_pre_wmma<<<gw, blk, 0, stream>>>(h1, W2, dinv, g, acc, n_tiles);
  }
  int rem = N - n_tiles * 16;
  if (rem > 0) {
    k_conv2_pre_tail<<<(rem + 63) / 64, 64, 0, stream>>>(h1, W2, dinv, g, acc, N,
                                                         n_tiles * 16);
  }

  // conv2 aggregate + head
  k_scatter<<<gs, blk, 0, stream>>>(src, dst, g, acc, E);
  k_final<<<gn, blk, 0, stream>>>(acc, dinv, b2, Wf, bf, out, N);
}